// SAGE_delta_7146825581285
// MI455X (gfx1250) — compile-verified
//
#include <hip/hip_runtime.h>

#define N_NODES 10000
#define D_H     512
#define N_CLS   64
#define N_EDGES 160000

typedef float v2f __attribute__((ext_vector_type(2)));
typedef float v8f __attribute__((ext_vector_type(8)));

// ---------------------------------------------------------------- utilities

__global__ void zero_f32(float* __restrict__ p, int n) {
    int i = blockIdx.x * blockDim.x + threadIdx.x;
    if (i < n) p[i] = 0.0f;
}

__global__ void copy_f32(const float* __restrict__ s, float* __restrict__ d, int n) {
    int i = blockIdx.x * blockDim.x + threadIdx.x;
    if (i < n) d[i] = s[i];
}

// ---------------------------------------------------------------- degree

__global__ void degree_kernel(const int* __restrict__ dst, float* __restrict__ deg) {
    int i = blockIdx.x * blockDim.x + threadIdx.x;
    if (i < N_EDGES) atomicAdd(&deg[dst[i]], 1.0f);
}

__global__ void invdeg_kernel(const float* __restrict__ deg, float* __restrict__ inv) {
    int i = blockIdx.x * blockDim.x + threadIdx.x;
    if (i < N_NODES) inv[i] = 1.0f / fmaxf(deg[i], 1.0f);
}

// ---------------------------------------------------------------- edge scatter-add
// One thread per (edge, 4-float chunk). float4 gather (L2-resident feature
// table) + 4 global_atomic_add_f32 scatters into the L2-resident agg buffer.

__global__ void scatter_add_kernel(const float* __restrict__ feat,
                                   const int* __restrict__ src,
                                   const int* __restrict__ dst,
                                   float* __restrict__ agg,
                                   int d4, int total) {
    int i = blockIdx.x * blockDim.x + threadIdx.x;
    if (i >= total) return;
    int e = i / d4;
    int c = i - e * d4;
    int s = src[e];
    int t = dst[e];
    float4 v = reinterpret_cast<const float4*>(feat)[(size_t)s * d4 + c];
    float* out = agg + ((size_t)t * d4 + c) * 4;
    atomicAdd(out + 0, v.x);
    atomicAdd(out + 1, v.y);
    atomicAdd(out + 2, v.z);
    atomicAdd(out + 3, v.w);
}

// ---------------------------------------------------------------- fused SAGE GEMM
// Y[N,M] = relu?( X@Wself + (AGG*invdeg)@Wneigh + bias )
// Each wave32 owns a 32x64 output macro-tile: 2 row-subtiles x 4 col-subtiles
// of 16x16, 8 f32 accumulators. K stepped by 4; 16 chained
// v_wmma_f32_16x16x4_f32 per k-step across 8 independent accumulation chains.
// A fragments reused 4x (cols), B fragments reused 2x (rows):
// ~192 B of L2 traffic per WMMA vs ~512 B for a naive 16x16/wave tiling.
//
// A 16x4 f32 layout : lanes 0-15 rows M=0..15 K={0,1}; lanes 16-31 same rows K={2,3}
// B 4x16  f32 layout: lanes 0-15 cols N=0..15 K={0,1}; lanes 16-31 same cols K={2,3}
// C/D 16x16 f32     : VGPR r -> M=r (lanes 0-15) / M=r+8 (lanes 16-31), N=lane&15
//
// N=10000 is a multiple of 16 but not 32: the last row group's upper subtile
// is invalid uniformly across the wave -> loads clamped to row N-1, store
// skipped by a wave-uniform guard (EXEC stays all-ones around every WMMA).

__global__ __launch_bounds__(256) void sage_gemm_wmma(
        const float* __restrict__ X,
        const float* __restrict__ AGG,
        const float* __restrict__ invdeg,
        const float* __restrict__ Wself,
        const float* __restrict__ Wneigh,
        const float* __restrict__ bias,
        float* __restrict__ Y,
        int K, int M, int doRelu) {
    const int colGroups = M >> 6;                       // 64-wide col groups
    const int rowGroups = ((N_NODES / 16) + 1) >> 1;    // 313 groups of 32 rows
    const int wave = threadIdx.x >> 5;
    const int g    = blockIdx.x * 8 + wave;
    if (g >= rowGroups * colGroups) return;             // whole-wave guard

    const int rg = g / colGroups;
    const int cg = g - rg * colGroups;

    const int lane = threadIdx.x & 31;
    const int half = lane >> 4;                         // 0: K pair {0,1}, 1: {2,3}
    const int l    = lane & 15;

    const int row0  = rg * 32 + l;
    const int row1  = row0 + 16;
    const int row1c = (row1 < N_NODES) ? row1 : (N_NODES - 1);  // clamp (uniform per subtile)
    const int colB  = cg * 64 + l;

    const float sc0 = invdeg[row0];
    const float sc1 = invdeg[row1c];
    const float* xr0 = X   + (size_t)row0  * K;
    const float* ar0 = AGG + (size_t)row0  * K;
    const float* xr1 = X   + (size_t)row1c * K;
    const float* ar1 = AGG + (size_t)row1c * K;

    v8f acc[2][4] = {};

#pragma unroll 2
    for (int k = 0; k < K; k += 4) {
        const int kb = k + 2 * half;
        v2f as0 = *reinterpret_cast<const v2f*>(xr0 + kb);       // 8B aligned (kb even)
        v2f an0 = *reinterpret_cast<const v2f*>(ar0 + kb) * sc0; // fold mean scale
        v2f as1 = *reinterpret_cast<const v2f*>(xr1 + kb);
        v2f an1 = *reinterpret_cast<const v2f*>(ar1 + kb) * sc1;

        const float* ws = Wself  + (size_t)kb * M + colB;
        const float* wn = Wneigh + (size_t)kb * M + colB;

#pragma unroll
        for (int s = 0; s < 4; ++s) {
            v2f bs, bn;
            bs.x = ws[s * 16];          // row kb
            bs.y = ws[M + s * 16];      // row kb+1
            bn.x = wn[s * 16];
            bn.y = wn[M + s * 16];
            acc[0][s] = __builtin_amdgcn_wmma_f32_16x16x4_f32(
                    false, as0, false, bs, (short)0, acc[0][s], false, false);
            acc[0][s] = __builtin_amdgcn_wmma_f32_16x16x4_f32(
                    false, an0, false, bn, (short)0, acc[0][s], false, false);
            acc[1][s] = __builtin_amdgcn_wmma_f32_16x16x4_f32(
                    false, as1, false, bs, (short)0, acc[1][s], false, false);
            acc[1][s] = __builtin_amdgcn_wmma_f32_16x16x4_f32(
                    false, an1, false, bn, (short)0, acc[1][s], false, false);
        }
    }

    // epilogue: bias (+relu) and store; row-subtile validity is wave-uniform
#pragma unroll
    for (int rs = 0; rs < 2; ++rs) {
        const int rowTile = rg * 32 + rs * 16;
        if (rowTile + 16 <= N_NODES) {
#pragma unroll
            for (int s = 0; s < 4; ++s) {
                const int col = colB + s * 16;
                const float bc = bias[col];
#pragma unroll
                for (int r = 0; r < 8; ++r) {
                    const int ro = rowTile + r + 8 * half;
                    float v = acc[rs][s][r] + bc;
                    if (doRelu) v = fmaxf(v, 0.0f);
                    Y[(size_t)ro * M + col] = v;
                }
            }
        }
    }
}

// ---------------------------------------------------------------- combine_embedding

__global__ void set_rows_kernel(float* __restrict__ out, const float* __restrict__ feat,
                                const int* __restrict__ idx, int nIdx) {
    int i = blockIdx.x * blockDim.x + threadIdx.x;
    if (i >= nIdx * N_CLS) return;
    int r = i / N_CLS, c = i - r * N_CLS;
    int node = idx[r];
    out[(size_t)node * N_CLS + c] = feat[(size_t)node * N_CLS + c];
}

__global__ void add_rows_kernel(float* __restrict__ out, const float* __restrict__ feat,
                                const int* __restrict__ idx, int nIdx) {
    int i = blockIdx.x * blockDim.x + threadIdx.x;
    if (i >= nIdx * N_CLS) return;
    int r = i / N_CLS, c = i - r * N_CLS;
    int node = idx[r];
    atomicAdd(&out[(size_t)node * N_CLS + c], feat[(size_t)node * N_CLS + c]);
}

// ---------------------------------------------------------------- launch

static inline int cdiv(int a, int b) { return (a + b - 1) / b; }

extern "C" void kernel_launch(void* const* d_in, const int* in_sizes, int n_in,
                              void* d_out, int out_size, void* d_ws, size_t ws_size,
                              hipStream_t stream) {
    const float* features = (const float*)d_in[0];
    const float* w_self0  = (const float*)d_in[1];
    const float* w_neigh0 = (const float*)d_in[2];
    const float* b0       = (const float*)d_in[3];
    const float* w_self1  = (const float*)d_in[4];
    const float* w_neigh1 = (const float*)d_in[5];
    const float* b1       = (const float*)d_in[6];
    const float* w_self2  = (const float*)d_in[7];
    const float* w_neigh2 = (const float*)d_in[8];
    const float* b2       = (const float*)d_in[9];
    const float* emb_prev = (const float*)d_in[10];
    const int*   edge_src = (const int*)d_in[11];
    const int*   edge_dst = (const int*)d_in[12];
    const int*   high_deg = (const int*)d_in[13];
    const int*   low_deg  = (const int*)d_in[14];
    float* out = (float*)d_out;

    // workspace layout (floats)
    float* ws   = (float*)d_ws;
    float* deg  = ws;                            // N
    float* invd = deg  + N_NODES;                // N
    float* agg  = invd + N_NODES;                // N * 512 (reused per layer)
    float* h1   = agg  + (size_t)N_NODES * D_H;  // N * 512
    float* h2   = h1   + (size_t)N_NODES * D_H;  // N * 512
    float* f3   = h2   + (size_t)N_NODES * D_H;  // N * 64

    const int B = 256;

    // degrees (shared by all 3 layers)
    zero_f32<<<cdiv(N_NODES, B), B, 0, stream>>>(deg, N_NODES);
    degree_kernel<<<cdiv(N_EDGES, B), B, 0, stream>>>(edge_dst, deg);
    invdeg_kernel<<<cdiv(N_NODES, B), B, 0, stream>>>(deg, invd);

    const int rowGroups = ((N_NODES / 16) + 1) / 2;           // 313
    const int waves512  = rowGroups * (D_H / 64);             // 2504 waves
    const int waves64   = rowGroups * (N_CLS / 64);           // 313 waves

    // ---- layer 0: features(512) -> h1(512), relu
    {
        int d4 = D_H / 4, total = N_EDGES * d4;
        zero_f32<<<cdiv(N_NODES * D_H, B), B, 0, stream>>>(agg, N_NODES * D_H);
        scatter_add_kernel<<<cdiv(total, B), B, 0, stream>>>(features, edge_src, edge_dst, agg, d4, total);
        sage_gemm_wmma<<<cdiv(waves512, 8), B, 0, stream>>>(
            features, agg, invd, w_self0, w_neigh0, b0, h1, D_H, D_H, 1);
    }
    // ---- layer 1: h1(512) -> h2(512), relu
    {
        int d4 = D_H / 4, total = N_EDGES * d4;
        zero_f32<<<cdiv(N_NODES * D_H, B), B, 0, stream>>>(agg, N_NODES * D_H);
        scatter_add_kernel<<<cdiv(total, B), B, 0, stream>>>(h1, edge_src, edge_dst, agg, d4, total);
        sage_gemm_wmma<<<cdiv(waves512, 8), B, 0, stream>>>(
            h1, agg, invd, w_self1, w_neigh1, b1, h2, D_H, D_H, 1);
    }
    // ---- layer 2: h2(512) -> f3(64), no relu
    {
        int d4 = N_CLS / 4, total = N_EDGES * d4;
        zero_f32<<<cdiv(N_NODES * N_CLS, B), B, 0, stream>>>(agg, N_NODES * N_CLS);
        scatter_add_kernel<<<cdiv(total, B), B, 0, stream>>>(h2, edge_src, edge_dst, agg, d4, total);
        sage_gemm_wmma<<<cdiv(waves64, 8), B, 0, stream>>>(
            h2, agg, invd, w_self2, w_neigh2, b2, f3, D_H, N_CLS, 0);
    }

    // ---- combine_embedding (ordered on stream; duplicates handled like JAX .at[].add)
    copy_f32<<<cdiv(N_NODES * N_CLS, B), B, 0, stream>>>(emb_prev, out, N_NODES * N_CLS);
    set_rows_kernel<<<cdiv(500 * N_CLS, B), B, 0, stream>>>(out, f3, high_deg, 500);
    set_rows_kernel<<<cdiv(500 * N_CLS, B), B, 0, stream>>>(out, f3, low_deg, 500);
    add_rows_kernel<<<cdiv(500 * N_CLS, B), B, 0, stream>>>(out, f3, low_deg, 500);
}